// DC1_14035953123414
// MI455X (gfx1250) — compile-verified
//
#include <hip/hip_runtime.h>
#include <math.h>

// ---------------------------------------------------------------------------
// CG solve of (A^H A + lam) x = atbT + lam*z, A^H A = ifft2(mask * fft2(.))
// FFTs realized as fp32 WMMA DFT matmuls:  fft2(X) = F * X * F  (F symmetric)
// ---------------------------------------------------------------------------

typedef __attribute__((ext_vector_type(2))) float v2f;
typedef __attribute__((ext_vector_type(8))) float v8f;

#define LAM     0.05f
#define CG_TOL  1e-10f
#define NB      16
#define NN      512
#define NPIX    (NB * NN * NN)     // complex elements per field

#define RED_BLOCKS  1024
#define RED_THREADS 256

// ------------------------------ reductions ---------------------------------

__device__ inline float2 blockReduce2(float vx, float vy) {
    __shared__ float sx[RED_THREADS];
    __shared__ float sy[RED_THREADS];
    int t = threadIdx.x;
    sx[t] = vx; sy[t] = vy;
    __syncthreads();
    for (int off = RED_THREADS >> 1; off > 0; off >>= 1) {
        if (t < off) { sx[t] += sx[t + off]; sy[t] += sy[t + off]; }
        __syncthreads();
    }
    return make_float2(sx[0], sy[0]);
}

__global__ __launch_bounds__(RED_THREADS)
void final_reduce(const float2* __restrict__ partials, float* __restrict__ dst) {
    float ax = 0.f, ay = 0.f;
    for (int i = threadIdx.x; i < RED_BLOCKS; i += RED_THREADS) {
        ax += partials[i].x; ay += partials[i].y;
    }
    float2 s = blockReduce2(ax, ay);
    if (threadIdx.x == 0) { dst[0] = s.x; dst[1] = s.y; }
}

// ------------------------------ twiddles -----------------------------------

__global__ void build_twiddles(float2* __restrict__ F, float2* __restrict__ Fi) {
    int idx = blockIdx.x * blockDim.x + threadIdx.x;
    if (idx >= NN * NN) return;
    int k = idx >> 9;
    int n = idx & (NN - 1);
    int t = (k * n) & (NN - 1);            // exact: e^{-2pi i kn/N} periodic in N
    float ang = (6.283185307179586f / (float)NN) * (float)t;
    float s, c;
    __sincosf(ang, &s, &c);
    F[idx]  = make_float2(c, -s);                              // forward DFT
    Fi[idx] = make_float2(c * (1.0f / NN), s * (1.0f / NN));   // inverse incl. 1/N
}

// -------------------------- complex WMMA GEMM ------------------------------
// C[b] = A[b?] (512x512 cx) * B[b?] (512x512 cx); optional real mask on output.
// Block: 256 threads = 8 waves as 2(M) x 4(N); each wave owns a 16x32 tile
// (two 16x16 WMMA tiles sharing one A fragment per K-step).
// Fragment layouts per CDNA5 ISA 7.12.2 (16x16x4 f32):
//   A: lane holds row M=lane&15, K pair selected by lane>>4  -> v2f {K0,K0+1}
//   B: lane holds col N=lane&15, rows K0,K0+1 in .x/.y
//   C: 8 VGPRs, element (M = v + 8*(lane>>4), N = lane&15)

__global__ __launch_bounds__(256)
void cgemm_tw(const float2* __restrict__ Abase, long aStride,
              const float2* __restrict__ Bbase, long bStride,
              float2* __restrict__ Cbase,
              const float* __restrict__ maskp)
{
    const int b = blockIdx.z;
    const float2* A  = Abase + (long)b * aStride;
    const float2* Bm = Bbase + (long)b * bStride;
    float2*       C  = Cbase + (long)b * (long)NN * NN;

    const int tid  = threadIdx.x;
    const int wave = tid >> 5;
    const int lane = tid & 31;
    const int mw = wave >> 2, nw = wave & 3;
    const int m0 = blockIdx.y * 32 + mw * 16;
    const int n0 = blockIdx.x * 128 + nw * 32;    // wave covers n0 .. n0+31

    const int row = lane & 15;
    const int kh  = lane >> 4;      // 0 or 1 -> K half
    const int col = lane & 15;

    v8f cr0 = {}, ci0 = {};
    v8f cr1 = {}, ci1 = {};

    const float2* arow  = A  + (long)(m0 + row) * NN;   // + k
    const float2* bcol0 = Bm + (n0 + col);              // + k*NN
    const float2* bcol1 = Bm + (n0 + 16 + col);         // + k*NN

    for (int kk = 0; kk < NN; kk += 4) {
        if ((kk & 63) == 0) {
            __builtin_prefetch((const void*)(arow + kk + 64), 0, 3);
            __builtin_prefetch((const void*)(bcol0 + (long)(kk + 64) * NN), 0, 3);
        }
        const int k0 = kk + kh * 2;

        // A fragment (complex pair for K0, K0+1) in one 16B load, shared by
        // both N tiles.
        float4 av = *(const float4*)(arow + k0);
        v2f ar  = { av.x,  av.z };
        v2f ai  = { av.y,  av.w };
        v2f nai = { -av.y, -av.w };   // f32 WMMA has no A-negate (NEG=C only)

        // B fragments for the two 16-column tiles
        float2 b00 = bcol0[(long)k0 * NN];
        float2 b01 = bcol0[(long)(k0 + 1) * NN];
        float2 b10 = bcol1[(long)k0 * NN];
        float2 b11 = bcol1[(long)(k0 + 1) * NN];
        v2f br0 = { b00.x, b01.x };
        v2f bi0 = { b00.y, b01.y };
        v2f br1 = { b10.x, b11.x };
        v2f bi1 = { b10.y, b11.y };

        // Cr += Ar*Br - Ai*Bi ; Ci += Ar*Bi + Ai*Br  (x2 tiles)
        cr0 = __builtin_amdgcn_wmma_f32_16x16x4_f32(false, ar,  false, br0, (short)0, cr0, false, false);
        cr0 = __builtin_amdgcn_wmma_f32_16x16x4_f32(false, nai, false, bi0, (short)0, cr0, false, false);
        ci0 = __builtin_amdgcn_wmma_f32_16x16x4_f32(false, ar,  false, bi0, (short)0, ci0, false, false);
        ci0 = __builtin_amdgcn_wmma_f32_16x16x4_f32(false, ai,  false, br0, (short)0, ci0, false, false);

        cr1 = __builtin_amdgcn_wmma_f32_16x16x4_f32(false, ar,  false, br1, (short)0, cr1, false, false);
        cr1 = __builtin_amdgcn_wmma_f32_16x16x4_f32(false, nai, false, bi1, (short)0, cr1, false, false);
        ci1 = __builtin_amdgcn_wmma_f32_16x16x4_f32(false, ar,  false, bi1, (short)0, ci1, false, false);
        ci1 = __builtin_amdgcn_wmma_f32_16x16x4_f32(false, ai,  false, br1, (short)0, ci1, false, false);
    }

    const int gn0 = n0 + col;
    const int gn1 = n0 + 16 + col;
    #pragma unroll
    for (int v = 0; v < 8; ++v) {
        const int m = m0 + 8 * kh + v;
        const float s0 = maskp ? maskp[m * NN + gn0] : 1.0f;
        const float s1 = maskp ? maskp[m * NN + gn1] : 1.0f;
        C[(long)m * NN + gn0] = make_float2(cr0[v] * s0, ci0[v] * s0);
        C[(long)m * NN + gn1] = make_float2(cr1[v] * s1, ci1[v] * s1);
    }
}

// --------------------------- CG pointwise kernels --------------------------

// r = p = atbT + lam*z ; x = 0 ; partial sums of |r|^2
__global__ __launch_bounds__(RED_THREADS)
void init_rp(const float2* __restrict__ z, const float2* __restrict__ atbT,
             float2* __restrict__ r, float2* __restrict__ p,
             float2* __restrict__ x, float2* __restrict__ partials)
{
    float acc = 0.f;
    const int stride = gridDim.x * blockDim.x;
    for (int i = blockIdx.x * blockDim.x + threadIdx.x; i < NPIX; i += stride) {
        float2 a = atbT[i];
        float2 zz = z[i];
        float2 rr = make_float2(a.x + LAM * zz.x, a.y + LAM * zz.y);
        r[i] = rr; p[i] = rr;
        x[i] = make_float2(0.f, 0.f);
        acc += rr.x * rr.x + rr.y * rr.y;
    }
    float2 s = blockReduce2(acc, 0.f);
    if (threadIdx.x == 0) partials[blockIdx.x] = s;
}

// ap += lam*p (in place) ; partial sums of <p, ap> (complex)
__global__ __launch_bounds__(RED_THREADS)
void fix_ap_dot(const float2* __restrict__ p, float2* __restrict__ ap,
                float2* __restrict__ partials)
{
    float accr = 0.f, acci = 0.f;
    const int stride = gridDim.x * blockDim.x;
    for (int i = blockIdx.x * blockDim.x + threadIdx.x; i < NPIX; i += stride) {
        float2 pp = p[i];
        float2 a  = ap[i];
        a.x += LAM * pp.x; a.y += LAM * pp.y;
        ap[i] = a;
        accr += pp.x * a.x + pp.y * a.y;   // Re(conj(p)*ap)
        acci += pp.x * a.y - pp.y * a.x;   // Im(conj(p)*ap)
    }
    float2 s = blockReduce2(accr, acci);
    if (threadIdx.x == 0) partials[blockIdx.x] = s;
}

// sc: [0]=rTr [1]=pAp.re [2]=pAp.im [3]=alpha.re [4]=alpha.im [5]=beta
//     [6]=active [8]=rTrNew
__global__ void scalarA(float* sc) {
    float rtr = sc[0];
    int   active = fabsf(rtr) > CG_TOL;
    float dr = active ? sc[1] : 1.0f;
    float di = active ? sc[2] : 0.0f;
    float d2 = dr * dr + di * di;
    float ar = rtr * dr / d2;
    float ai = -rtr * di / d2;
    sc[3] = active ? ar : 0.0f;   // alpha applied only when active
    sc[4] = active ? ai : 0.0f;
    sc[6] = active ? 1.0f : 0.0f;
}

// x += alpha*p ; r -= alpha*ap ; partial sums of |r_new|^2
__global__ __launch_bounds__(RED_THREADS)
void update_xr(const float2* __restrict__ p, const float2* __restrict__ ap,
               float2* __restrict__ x, float2* __restrict__ r,
               const float* __restrict__ sc, float2* __restrict__ partials)
{
    const float ar = sc[3];
    const float ai = sc[4];
    float acc = 0.f;
    const int stride = gridDim.x * blockDim.x;
    for (int i = blockIdx.x * blockDim.x + threadIdx.x; i < NPIX; i += stride) {
        float2 pp = p[i];
        float2 aa = ap[i];
        float2 xx = x[i];
        float2 rr = r[i];
        xx.x += ar * pp.x - ai * pp.y;
        xx.y += ar * pp.y + ai * pp.x;
        rr.x -= ar * aa.x - ai * aa.y;
        rr.y -= ar * aa.y + ai * aa.x;
        x[i] = xx; r[i] = rr;
        acc += rr.x * rr.x + rr.y * rr.y;
    }
    float2 s = blockReduce2(acc, 0.f);
    if (threadIdx.x == 0) partials[blockIdx.x] = s;
}

__global__ void scalarB(float* sc) {
    float act    = sc[6];
    float rtrNew = sc[8];
    float rtrOld = sc[0];
    float safe   = (act > 0.5f) ? rtrOld : 1.0f;
    sc[5] = (act > 0.5f) ? (rtrNew / safe) : 0.0f;
    if (act > 0.5f) sc[0] = rtrNew;
}

// p = r + beta*p  (only when active)
__global__ __launch_bounds__(RED_THREADS)
void update_p(const float2* __restrict__ r, float2* __restrict__ p,
              const float* __restrict__ sc)
{
    const float act  = sc[6];
    const float beta = sc[5];
    const int stride = gridDim.x * blockDim.x;
    for (int i = blockIdx.x * blockDim.x + threadIdx.x; i < NPIX; i += stride) {
        float2 pp = p[i];
        float2 rr = r[i];
        float2 np = make_float2(rr.x + beta * pp.x, rr.y + beta * pp.y);
        p[i] = (act > 0.5f) ? np : pp;
    }
}

// ------------------------------- launcher ----------------------------------

extern "C" void kernel_launch(void* const* d_in, const int* in_sizes, int n_in,
                              void* d_out, int out_size, void* d_ws, size_t ws_size,
                              hipStream_t stream)
{
    (void)in_sizes; (void)n_in; (void)out_size; (void)ws_size;

    const float2* z    = (const float2*)d_in[0];
    const float2* atbT = (const float2*)d_in[1];
    const float*  mask = (const float*)d_in[2];
    float2* x = (float2*)d_out;

    char* w = (char*)d_ws;
    float2* F  = (float2*)w;  w += (size_t)NN * NN * sizeof(float2);
    float2* Fi = (float2*)w;  w += (size_t)NN * NN * sizeof(float2);
    float2* r  = (float2*)w;  w += (size_t)NPIX * sizeof(float2);
    float2* p  = (float2*)w;  w += (size_t)NPIX * sizeof(float2);
    float2* t1 = (float2*)w;  w += (size_t)NPIX * sizeof(float2);
    float2* t2 = (float2*)w;  w += (size_t)NPIX * sizeof(float2);   // also Ap
    float2* partials = (float2*)w; w += (size_t)RED_BLOCKS * sizeof(float2);
    float*  sc = (float*)w;

    const long fieldStride = (long)NN * NN;   // complex elems per batch image

    dim3 gemmGrid(NN / 128, NN / 32, NB);     // (4, 16, 16)
    dim3 gemmBlk(256);

    build_twiddles<<<(NN * NN + 255) / 256, 256, 0, stream>>>(F, Fi);
    init_rp<<<RED_BLOCKS, RED_THREADS, 0, stream>>>(z, atbT, r, p, x, partials);
    final_reduce<<<1, RED_THREADS, 0, stream>>>(partials, sc + 0);   // sc[0]=rTr

    for (int it = 0; it < 10; ++it) {
        // Ap_raw = ifft2(mask * fft2(p)) via four complex DFT GEMMs
        cgemm_tw<<<gemmGrid, gemmBlk, 0, stream>>>(p,  fieldStride, F,  0L,          t1, nullptr); // t1 = p * F
        cgemm_tw<<<gemmGrid, gemmBlk, 0, stream>>>(F,  0L,          t1, fieldStride, t2, mask);    // t2 = mask .* (F * t1)
        cgemm_tw<<<gemmGrid, gemmBlk, 0, stream>>>(t2, fieldStride, Fi, 0L,          t1, nullptr); // t1 = t2 * Fi
        cgemm_tw<<<gemmGrid, gemmBlk, 0, stream>>>(Fi, 0L,          t1, fieldStride, t2, nullptr); // t2 = Fi * t1

        // Ap = Ap_raw + lam*p ; pAp = <p, Ap>
        fix_ap_dot<<<RED_BLOCKS, RED_THREADS, 0, stream>>>(p, t2, partials);
        final_reduce<<<1, RED_THREADS, 0, stream>>>(partials, sc + 1);  // sc[1..2]=pAp

        scalarA<<<1, 1, 0, stream>>>(sc);

        update_xr<<<RED_BLOCKS, RED_THREADS, 0, stream>>>(p, t2, x, r, sc, partials);
        final_reduce<<<1, RED_THREADS, 0, stream>>>(partials, sc + 8);  // sc[8]=rTrNew

        scalarB<<<1, 1, 0, stream>>>(sc);

        update_p<<<RED_BLOCKS, RED_THREADS, 0, stream>>>(r, p, sc);
    }
}